// GCN_14328010899632
// MI455X (gfx1250) — compile-verified
//
#include <hip/hip_runtime.h>

typedef __attribute__((ext_vector_type(2))) float v2f;
typedef __attribute__((ext_vector_type(8))) float v8f;

#define NN 50000
#define NE 800000
#define MT (NN / 16)   // 3125 row-tiles, exact (50000 % 16 == 0)

// ---------------- init: deg = 1 (self loop), zero the aggregation buffers ---
__global__ __launch_bounds__(256)
void k_init(float* __restrict__ deg, float* __restrict__ agg1,
            float* __restrict__ agg2) {
  long stride = (long)gridDim.x * blockDim.x;
  long t0 = (long)blockIdx.x * blockDim.x + threadIdx.x;
  for (long i = t0; i < NN; i += stride)            deg[i]  = 1.0f;
  for (long i = t0; i < (long)NN * 64; i += stride) agg1[i] = 0.0f;
  for (long i = t0; i < (long)NN * 16; i += stride) agg2[i] = 0.0f;
}

// ---------------- degree count over dst ------------------------------------
__global__ __launch_bounds__(256)
void k_degree(const int* __restrict__ dst, float* __restrict__ deg) {
  int e = blockIdx.x * 256 + threadIdx.x;
  if (e < NE) unsafeAtomicAdd(&deg[dst[e]], 1.0f);
}

// ---------------- dinv = rsqrt(deg), in place ------------------------------
__global__ __launch_bounds__(256)
void k_rsqrt(float* __restrict__ deg) {
  int i = blockIdx.x * 256 + threadIdx.x;
  if (i < NN) deg[i] = rsqrtf(deg[i]);
}

// ---------------- h1 = x @ W1  (50000x64 @ 64x64), fp32 WMMA ---------------
// One 16-row tile per wave; W1 (16 KB) staged in LDS; 64 wmma/wave.
__global__ __launch_bounds__(256)
void k_gemm64(const float* __restrict__ x, const float* __restrict__ W,
              float* __restrict__ h) {
  __shared__ float lw[64 * 64];
  int t = threadIdx.x;
#pragma unroll
  for (int i = 0; i < 16; ++i) lw[t + i * 256] = W[t + i * 256];
  __syncthreads();

  int lane = t & 31;
  int tile = blockIdx.x * 8 + (t >> 5);
  if (tile >= MT) return;                 // wave-uniform branch, EXEC stays full

  int m  = lane & 15;                     // A row / B col within tile
  int kh = (lane >> 4) << 1;              // 0 or 2 (K half per ISA A-layout)
  const float* xr = x + ((long)tile * 16 + m) * 64;

  v8f c0 = {}, c1 = {}, c2 = {}, c3 = {};
#pragma unroll
  for (int k0 = 0; k0 < 64; k0 += 4) {
    v2f a, b0, b1, b2, b3;
    a.x = xr[k0 + kh];
    a.y = xr[k0 + kh + 1];
    const float* wr = &lw[(k0 + kh) * 64 + m];
    b0.x = wr[0];  b0.y = wr[64];
    b1.x = wr[16]; b1.y = wr[80];
    b2.x = wr[32]; b2.y = wr[96];
    b3.x = wr[48]; b3.y = wr[112];
    c0 = __builtin_amdgcn_wmma_f32_16x16x4_f32(false, a, false, b0, (short)0, c0, false, false);
    c1 = __builtin_amdgcn_wmma_f32_16x16x4_f32(false, a, false, b1, (short)0, c1, false, false);
    c2 = __builtin_amdgcn_wmma_f32_16x16x4_f32(false, a, false, b2, (short)0, c2, false, false);
    c3 = __builtin_amdgcn_wmma_f32_16x16x4_f32(false, a, false, b3, (short)0, c3, false, false);
  }

  float* ho = h + (long)tile * 16 * 64;
  int n  = lane & 15;
  int mb = (lane >> 4) << 3;              // C/D: VGPR v holds M = v + 8*(lane>=16)
#pragma unroll
  for (int v = 0; v < 8; ++v) {
    long r = (long)(mb + v) * 64;
    ho[r + n]      = c0[v];
    ho[r + n + 16] = c1[v];
    ho[r + n + 32] = c2[v];
    ho[r + n + 48] = c3[v];
  }
}

// ---------------- layer-1 edge scatter: agg1[dst] += h1[src]*norm ----------
__global__ __launch_bounds__(256)
void k_scatter64(const int* __restrict__ src, const int* __restrict__ dst,
                 const float* __restrict__ dinv, const float* __restrict__ h,
                 float* __restrict__ agg) {
  long idx = (long)blockIdx.x * 256 + threadIdx.x;   // E*16 threads
  if (idx >= (long)NE * 16) return;
  int e = (int)(idx >> 4);
  int q = (int)(idx & 15);
  int s = src[e], d = dst[e];
  float norm = dinv[s] * dinv[d];
  const float4* hv = (const float4*)(h + (long)s * 64);
  float4 mv = hv[q];
  float* a = agg + (long)d * 64 + q * 4;
  unsafeAtomicAdd(a + 0, mv.x * norm);
  unsafeAtomicAdd(a + 1, mv.y * norm);
  unsafeAtomicAdd(a + 2, mv.z * norm);
  unsafeAtomicAdd(a + 3, mv.w * norm);
}

// ---------------- self-loop + bias + ReLU (in place into agg1) -------------
__global__ __launch_bounds__(256)
void k_self_relu(const float* __restrict__ h, const float* __restrict__ dinv,
                 const float* __restrict__ b, float* __restrict__ agg) {
  long idx = (long)blockIdx.x * 256 + threadIdx.x;
  if (idx >= (long)NN * 64) return;
  int i = (int)(idx >> 6), f = (int)(idx & 63);
  float di = dinv[i];
  float v = agg[idx] + h[idx] * di * di + b[f];
  agg[idx] = fmaxf(v, 0.0f);
}

// ---------------- h2 = relu(h1') @ W2  (50000x64 @ 64x16), fp32 WMMA -------
__global__ __launch_bounds__(256)
void k_gemm16(const float* __restrict__ hr, const float* __restrict__ W,
              float* __restrict__ h2) {
  __shared__ float lw[64 * 16];
  int t = threadIdx.x;
#pragma unroll
  for (int i = 0; i < 4; ++i) lw[t + i * 256] = W[t + i * 256];
  __syncthreads();

  int lane = t & 31;
  int tile = blockIdx.x * 8 + (t >> 5);
  if (tile >= MT) return;

  int n  = lane & 15;
  int kh = (lane >> 4) << 1;
  const float* xr = hr + ((long)tile * 16 + n) * 64;

  v8f c = {};
#pragma unroll
  for (int k0 = 0; k0 < 64; k0 += 4) {
    v2f a, b;
    a.x = xr[k0 + kh];
    a.y = xr[k0 + kh + 1];
    b.x = lw[(k0 + kh) * 16 + n];
    b.y = lw[(k0 + kh + 1) * 16 + n];
    c = __builtin_amdgcn_wmma_f32_16x16x4_f32(false, a, false, b, (short)0, c, false, false);
  }

  float* ho = h2 + (long)tile * 16 * 16;
  int mb = (lane >> 4) << 3;
#pragma unroll
  for (int v = 0; v < 8; ++v) ho[(long)(mb + v) * 16 + n] = c[v];
}

// ---------------- layer-2 edge scatter -------------------------------------
__global__ __launch_bounds__(256)
void k_scatter16(const int* __restrict__ src, const int* __restrict__ dst,
                 const float* __restrict__ dinv, const float* __restrict__ h,
                 float* __restrict__ agg) {
  long idx = (long)blockIdx.x * 256 + threadIdx.x;   // E*4 threads
  if (idx >= (long)NE * 4) return;
  int e = (int)(idx >> 2);
  int q = (int)(idx & 3);
  int s = src[e], d = dst[e];
  float norm = dinv[s] * dinv[d];
  const float4* hv = (const float4*)(h + (long)s * 16);
  float4 mv = hv[q];
  float* a = agg + (long)d * 16 + q * 4;
  unsafeAtomicAdd(a + 0, mv.x * norm);
  unsafeAtomicAdd(a + 1, mv.y * norm);
  unsafeAtomicAdd(a + 2, mv.z * norm);
  unsafeAtomicAdd(a + 3, mv.w * norm);
}

// ---------------- self-loop + bias + sigmoid -> out ------------------------
__global__ __launch_bounds__(256)
void k_final(const float* __restrict__ h2, const float* __restrict__ dinv,
             const float* __restrict__ b, const float* __restrict__ agg,
             float* __restrict__ out) {
  long idx = (long)blockIdx.x * 256 + threadIdx.x;
  if (idx >= (long)NN * 16) return;
  int i = (int)(idx >> 4), f = (int)(idx & 15);
  float di = dinv[i];
  float v = agg[idx] + h2[idx] * di * di + b[f];
  out[idx] = 1.0f / (1.0f + __expf(-v));
}

extern "C" void kernel_launch(void* const* d_in, const int* in_sizes, int n_in,
                              void* d_out, int out_size, void* d_ws, size_t ws_size,
                              hipStream_t stream) {
  const float* x  = (const float*)d_in[0];
  const int*   ei = (const int*)d_in[1];      // [2, E] flat: src then dst
  const float* W1 = (const float*)d_in[2];
  const float* b1 = (const float*)d_in[3];
  const float* W2 = (const float*)d_in[4];
  const float* b2 = (const float*)d_in[5];
  float* out = (float*)d_out;

  float* ws   = (float*)d_ws;                 // 161*N floats ~= 32.2 MB
  float* deg  = ws;                           // N   (becomes dinv in place)
  float* h1   = deg  + NN;                    // N*64
  float* agg1 = h1   + (size_t)NN * 64;       // N*64 (becomes relu(h1') in place)
  float* h2   = agg1 + (size_t)NN * 64;       // N*16
  float* agg2 = h2   + (size_t)NN * 16;       // N*16

  const int* srcp = ei;
  const int* dstp = ei + NE;

  k_init     <<<2048,            256, 0, stream>>>(deg, agg1, agg2);
  k_degree   <<<(NE + 255)/256,  256, 0, stream>>>(dstp, deg);
  k_rsqrt    <<<(NN + 255)/256,  256, 0, stream>>>(deg);
  k_gemm64   <<<(MT + 7)/8,      256, 0, stream>>>(x, W1, h1);
  k_scatter64<<<(NE*16)/256,     256, 0, stream>>>(srcp, dstp, deg, h1, agg1);
  k_self_relu<<<(NN*64)/256,     256, 0, stream>>>(h1, deg, b1, agg1);
  k_gemm16   <<<(MT + 7)/8,      256, 0, stream>>>(agg1, W2, h2);
  k_scatter16<<<(NE*4)/256,      256, 0, stream>>>(srcp, dstp, deg, h2, agg2);
  k_final    <<<(NN*16)/256,     256, 0, stream>>>(h2, deg, b2, agg2, out);
}